// VariationalMPS_36438502539702
// MI455X (gfx1250) — compile-verified
//
#include <hip/hip_runtime.h>

// VariationalMPS on gfx1250: fp32 WMMA (16x16x4) sweep.
// Layouts (all float32, k-index fastest for both GEMM operands):
//   Lbuf[w][b][a]     = L(a,w,b)                (8*128*128)
//   Nbuf[q][p]        = N'(p,q)                 (128*128)
//   Tbuf[w][b][c]     = T(w,b,i,p), c=i*128+p   (8*128*256)
//   T2buf[x][p][r]    = T2(b,p,x,j), r=j*128+b  (8*128*256)
//   Tv2[p][s]         = T(b,i,p),  s=i*128+b    (128*256)
//   MT1[c][a]         = M[a][i][p], c=i*128+p   (256*128)  per-site transpose
//   MT3[q][s]         = M[b][i][q], s=i*128+b   (128*256)  per-site transpose

typedef __attribute__((ext_vector_type(2))) float v2f;
typedef __attribute__((ext_vector_type(8))) float v8f;

__device__ __forceinline__ v8f wmma_f32(v2f a, v2f b, v8f c) {
  // D = A(16x4) * B(4x16) + C(16x16), fp32 throughout (matches reference dtype)
  return __builtin_amdgcn_wmma_f32_16x16x4_f32(false, a, false, b, (short)0, c,
                                               false, false);
}

// C[m0:m0+16, n0:n0+64] = sum_k SA[m][k] * SB[n][k]^T-as-B, K-contiguous b64 loads.
__device__ __forceinline__ void gemm16x64(const float* __restrict__ SA, int lda,
                                          const float* __restrict__ SB, int ldb,
                                          int m0, int n0, int K, v8f acc[4]) {
  const int lane = threadIdx.x & 31;
  const int half = lane >> 4;
  const int l = lane & 15;
  const float* pa = SA + (m0 + l) * lda + 2 * half;
  const float* pb = SB + (n0 + l) * ldb + 2 * half;
  v8f c0 = {0.f, 0.f, 0.f, 0.f, 0.f, 0.f, 0.f, 0.f};
  v8f c1 = c0, c2 = c0, c3 = c0;
#pragma unroll 4
  for (int k = 0; k < K; k += 4) {
    v2f a  = *(const v2f*)(pa + k);
    v2f b0 = *(const v2f*)(pb + k);
    v2f b1 = *(const v2f*)(pb + 16 * ldb + k);
    v2f b2 = *(const v2f*)(pb + 32 * ldb + k);
    v2f b3 = *(const v2f*)(pb + 48 * ldb + k);
    c0 = wmma_f32(a, b0, c0);
    c1 = wmma_f32(a, b1, c1);
    c2 = wmma_f32(a, b2, c2);
    c3 = wmma_f32(a, b3, c3);
  }
  acc[0] = c0; acc[1] = c1; acc[2] = c2; acc[3] = c3;
}

// ---------------- init: L0, N0 ----------------
__global__ __launch_bounds__(256) void kInit(float* Lbuf, float* Nbuf) {
  int tid = blockIdx.x * 256 + threadIdx.x;
  if (tid < 8 * 128 * 128) Lbuf[tid] = (tid == 0) ? 1.0f : 0.0f;  // L(0,0,0)=1
  if (tid < 128 * 128)     Nbuf[tid] = (tid == 0) ? 1.0f : 0.0f;  // N(0,0)=1
}

// ---------------- per-site transposes of M ----------------
__global__ __launch_bounds__(256) void kT(const float* __restrict__ Msite,
                                          float* __restrict__ MT1,
                                          float* __restrict__ MT3) {
  int tid = blockIdx.x * 256 + threadIdx.x;   // 0..32767
  int a = tid >> 8, i = (tid >> 7) & 1, p = tid & 127;
  float v = Msite[tid];                        // M[a][i][p] (coalesced read)
  MT1[(i * 128 + p) * 128 + a] = v;            // [c][a]
  MT3[p * 256 + i * 128 + a]   = v;            // [q][s], s=i*128+b
}

// ---------------- E-step1 (8 GEMMs) + N-step1 (1 GEMM) ----------------
__global__ __launch_bounds__(256) void kB(const float* __restrict__ Lbuf,
                                          const float* __restrict__ MT1,
                                          const float* __restrict__ Nbuf,
                                          float* __restrict__ Tbuf,
                                          float* __restrict__ Tv2) {
  int wgid = blockIdx.x * 8 + (threadIdx.x >> 5);  // 0..287
  int lane = threadIdx.x & 31, half = lane >> 4, l = lane & 15;
  v8f acc[4];
  if (wgid < 256) {
    // T_w(b,c) = sum_a Lbuf[w][b][a] * MT1[c][a]
    int w = wgid >> 5, rem = wgid & 31;
    int m0 = (rem >> 2) * 16;   // b tile
    int n0 = (rem & 3) * 64;    // c group
    gemm16x64(Lbuf + w * 128 * 128, 128, MT1, 128, m0, n0, 128, acc);
    float* C = Tbuf + w * 128 * 256;
#pragma unroll
    for (int t = 0; t < 4; ++t)
#pragma unroll
      for (int v = 0; v < 8; ++v)
        C[(m0 + v + 8 * half) * 256 + n0 + 16 * t + l] = acc[t][v];
  } else {
    // T(b,i,p) = sum_a MT1[c][a] * Nbuf[b][a]   -> Tv2[p][i*128+b]
    int t2 = wgid - 256;        // 0..31
    int m0 = (t2 >> 1) * 16;    // c tile (0..255)
    int n0 = (t2 & 1) * 64;     // b group
    gemm16x64(MT1, 128, Nbuf, 128, m0, n0, 128, acc);
#pragma unroll
    for (int t = 0; t < 4; ++t)
#pragma unroll
      for (int v = 0; v < 8; ++v) {
        int c = m0 + v + 8 * half;
        int i = c >> 7, p = c & 127;
        Tv2[p * 256 + i * 128 + n0 + 16 * t + l] = acc[t][v];
      }
  }
}

// ---------------- E-step2: MPO contraction (VALU) ----------------
__global__ __launch_bounds__(256) void k2(const float* __restrict__ Tbuf,
                                          const float* __restrict__ Hsite,
                                          float* __restrict__ T2buf) {
  int tid = blockIdx.x * 256 + threadIdx.x;  // 0..16383
  int p = tid >> 7, b = tid & 127;
  float t[16];
#pragma unroll
  for (int w = 0; w < 8; ++w)
#pragma unroll
    for (int i = 0; i < 2; ++i)
      t[w * 2 + i] = Tbuf[(w * 128 + b) * 256 + i * 128 + p];
#pragma unroll
  for (int x = 0; x < 8; ++x) {
    float o0 = 0.f, o1 = 0.f;
#pragma unroll
    for (int w = 0; w < 8; ++w)
#pragma unroll
      for (int i = 0; i < 2; ++i) {
        float tv = t[w * 2 + i];
        o0 += tv * Hsite[((w * 8 + x) * 2 + i) * 2 + 0];
        o1 += tv * Hsite[((w * 8 + x) * 2 + i) * 2 + 1];
      }
    T2buf[(x * 128 + p) * 256 + 0 * 128 + b] = o0;  // r = j*128+b
    T2buf[(x * 128 + p) * 256 + 1 * 128 + b] = o1;
  }
}

// ---------------- E-step3 (8 GEMMs) + N-step2 (1 GEMM) ----------------
__global__ __launch_bounds__(256) void kD(const float* __restrict__ MT3,
                                          const float* __restrict__ T2buf,
                                          const float* __restrict__ Tv2,
                                          float* __restrict__ Lbuf,
                                          float* __restrict__ Nbuf) {
  int wgid = blockIdx.x * 8 + (threadIdx.x >> 5);  // 0..143
  int lane = threadIdx.x & 31, half = lane >> 4, l = lane & 15;
  v8f acc[4];
  const float* SB;
  float* C;
  int m0, n0;
  if (wgid < 128) {
    // L'(q,p)[x] = sum_r MT3[q][r] * T2buf[x][p][r]  -> Lbuf[x][q][p]
    int x = wgid >> 4, rem = wgid & 15;
    m0 = (rem >> 1) * 16;  // q tile
    n0 = (rem & 1) * 64;   // p group
    SB = T2buf + x * 128 * 256;
    C = Lbuf + x * 128 * 128;
  } else {
    // N'(q,p) = sum_s MT3[q][s] * Tv2[p][s]  -> Nbuf[q][p]
    int t2 = wgid - 128;
    m0 = (t2 >> 1) * 16;
    n0 = (t2 & 1) * 64;
    SB = Tv2;
    C = Nbuf;
  }
  gemm16x64(MT3, 256, SB, 256, m0, n0, 256, acc);
#pragma unroll
  for (int t = 0; t < 4; ++t)
#pragma unroll
    for (int v = 0; v < 8; ++v)
      C[(m0 + v + 8 * half) * 128 + n0 + 16 * t + l] = acc[t][v];
}

// ---------------- finalize ----------------
__global__ void kFin(const float* __restrict__ Lbuf,
                     const float* __restrict__ Nbuf, float* __restrict__ out) {
  if (threadIdx.x == 0 && blockIdx.x == 0) {
    float e = Lbuf[7 * 128 * 128];  // L'(p=0, x=W-1, q=0)
    float n = Nbuf[0];              // N'(0,0)
    out[0] = e;
    out[1] = n;
    out[2] = e / n;
    out[3] = fmaxf(n - 10000.0f, 0.0f);
  }
}

extern "C" void kernel_launch(void* const* d_in, const int* in_sizes, int n_in,
                              void* d_out, int out_size, void* d_ws,
                              size_t ws_size, hipStream_t stream) {
  (void)in_sizes; (void)n_in; (void)out_size; (void)ws_size;
  const float* M = (const float*)d_in[0];  // (40,128,2,128)
  const float* H = (const float*)d_in[1];  // (40,8,8,2,2)
  float* ws = (float*)d_ws;
  float* Lbuf  = ws;                  // 131072
  float* Nbuf  = Lbuf + 131072;       // 16384
  float* Tbuf  = Nbuf + 16384;        // 262144
  float* T2buf = Tbuf + 262144;       // 262144
  float* Tv2   = T2buf + 262144;      // 32768
  float* MT1   = Tv2 + 32768;         // 32768
  float* MT3   = MT1 + 32768;         // 32768  (total ~2.94 MB)

  kInit<<<512, 256, 0, stream>>>(Lbuf, Nbuf);
  for (int s = 0; s < 40; ++s) {
    kT<<<128, 256, 0, stream>>>(M + (size_t)s * 32768, MT1, MT3);
    kB<<<36, 256, 0, stream>>>(Lbuf, MT1, Nbuf, Tbuf, Tv2);
    k2<<<64, 256, 0, stream>>>(Tbuf, H + (size_t)s * 256, T2buf);
    kD<<<18, 256, 0, stream>>>(MT3, T2buf, Tv2, Lbuf, Nbuf);
  }
  kFin<<<1, 1, 0, stream>>>(Lbuf, Nbuf, (float*)d_out);
}